// TanBayesNetClassifierStructurePenalty_37194416783933
// MI455X (gfx1250) — compile-verified
//
#include <hip/hip_runtime.h>
#include <hip/hip_bf16.h>
#include <math.h>

// Problem constants (match reference)
#define NF   24          // F features
#define NU   64          // U values per feature
#define NC   128         // C classes
#define NP   276         // F*(F-1)/2 pairs
#define EPSV 1e-6f

// Async gather pipeline: 276 = 23 groups of 12 pairs, double buffered.
#define G_PAIRS  12
#define N_GROUPS 23

#if __has_builtin(__builtin_amdgcn_global_load_async_to_lds_b32)
#define HAVE_ASYNC_LDS 1
#else
#define HAVE_ASYNC_LDS 0
#endif

#if __has_builtin(__builtin_amdgcn_wmma_f32_16x16x4_f32)
#define HAVE_WMMA 1
#else
#define HAVE_WMMA 0
#endif

typedef __attribute__((address_space(1))) int global_int;
typedef __attribute__((address_space(3))) int lds_int;
typedef float v2f __attribute__((ext_vector_type(2)));
typedef float v8f __attribute__((ext_vector_type(8)));

// ---- CDNA5 wait helpers (ASYNCcnt / DScnt) -------------------------------
__device__ __forceinline__ void wait_dscnt0() {
  asm volatile("s_wait_dscnt 0" ::: "memory");
}
template <int N>
__device__ __forceinline__ void wait_asynccnt() {
  asm volatile("s_wait_asynccnt %0" :: "n"(N) : "memory");
}

#if HAVE_ASYNC_LDS
// One dword per lane: copied from the global address in gsrc to the LDS address in lds_dst, tracked by ASYNCcnt.
__device__ __forceinline__ void async_ld_f32(const float* gsrc, float* lds_dst) {
  __builtin_amdgcn_global_load_async_to_lds_b32(
      (global_int*)gsrc, (lds_int*)lds_dst, 0, 0);
}
#endif

// ---- Kernel 3 (listed first for disasm visibility): main gather ----------
// One block per batch element b, thread == class c. Pair rows are staged to
// LDS with async global->LDS loads, double buffered (24 in flight per wave).
__global__ void __launch_bounds__(128)
main_kernel(const int* __restrict__ x, const float* __restrict__ featL,
            const float* __restrict__ aug, const float* __restrict__ base,
            const float* __restrict__ fw, const float* __restrict__ pw,
            const float* __restrict__ lseT, float* __restrict__ out) {
  __shared__ int x_s[NF];
  __shared__ float fw_s[NF];
  __shared__ float pw_s[NP];
  __shared__ unsigned char pf_s[NP];
  __shared__ unsigned char pj_s[NP];
#if HAVE_ASYNC_LDS
  __shared__ float augbuf[2][G_PAIRS][NC];
  __shared__ float lsebuf[2][G_PAIRS][NC];
#endif

  const int c = threadIdx.x;
  const int b = blockIdx.x;

  if (c < NF) {
    x_s[c] = x[b * NF + c];
    fw_s[c] = fw[c];
  }
  for (int p = c; p < NP; p += NC) {
    pw_s[p] = pw[p];
    int f = 1;
    while ((f * (f + 1)) / 2 <= p) ++f;
    pf_s[p] = (unsigned char)f;
    pj_s[p] = (unsigned char)(p - (f * (f - 1)) / 2);
  }
  __syncthreads();

#if HAVE_ASYNC_LDS
  auto issue = [&](int gi, int buf) {
#pragma unroll
    for (int k = 0; k < G_PAIRS; ++k) {
      int p = gi * G_PAIRS + k;
      int f = pf_s[p], j = pj_s[p];
      size_t aidx = (((size_t)p * NU + x_s[f]) * NU + x_s[j]) * NC + c;
      size_t lidx = ((size_t)p * NU + x_s[j]) * NC + c;
      async_ld_f32(&aug[aidx], &augbuf[buf][k][c]);
      async_ld_f32(&lseT[lidx], &lsebuf[buf][k][c]);
    }
  };
  issue(0, 0);  // start first gather DMAs before the unary-feature work
#endif

  // Unary feature terms (feature table is L2-resident, 768KB) — overlaps the
  // in-flight async group.
  float acc = base[c] + featL[(size_t)x_s[0] * NC + c];
#pragma unroll
  for (int f = 1; f < NF; ++f)
    acc += fw_s[f] * featL[((size_t)f * NU + x_s[f]) * NC + c];

#if HAVE_ASYNC_LDS
  volatile const float* augv = &augbuf[0][0][0];
  volatile const float* lsev = &lsebuf[0][0][0];

  for (int gi = 0; gi < N_GROUPS; ++gi) {
    const int cur = gi & 1;
    if (gi + 1 < N_GROUPS) {
      wait_dscnt0();              // close DS-read vs async-write WAR on buffer
      issue(gi + 1, cur ^ 1);
      wait_asynccnt<2 * G_PAIRS>();  // in-order: group gi fully landed
    } else {
      wait_asynccnt<0>();
    }
#pragma unroll
    for (int k = 0; k < G_PAIRS; ++k) {
      int p = gi * G_PAIRS + k;
      float av = augv[(cur * G_PAIRS + k) * NC + c];
      float lv = lsev[(cur * G_PAIRS + k) * NC + c];
      acc += pw_s[p] * (av - lv);
    }
  }
#else
  // Fallback: direct coalesced gathers (still 512B rows, lane == c).
  for (int p = 0; p < NP; ++p) {
    int f = pf_s[p], j = pj_s[p];
    size_t aidx = (((size_t)p * NU + x_s[f]) * NU + x_s[j]) * NC + c;
    size_t lidx = ((size_t)p * NU + x_s[j]) * NC + c;
    acc += pw_s[p] * (aug[aidx] - lseT[lidx]);
  }
#endif

  out[(size_t)b * NC + c] = acc;
}

// ---- Kernel 2: lseAug[p,u2,c] = logsumexp_u1 aug[p,u1,u2,c] --------------
// Lane == c  => fully coalesced 512B lines per (p,u1,u2). Streams 579MB once.
__global__ void __launch_bounds__(256)
lse_aug_kernel(const float* __restrict__ aug, float* __restrict__ lseT) {
  size_t i = (size_t)blockIdx.x * 256 + threadIdx.x;
  if (i >= (size_t)NP * NU * NC) return;
  size_t p = i >> 13;         // / (64*128)
  size_t u2c = i & 8191;      // u2*128 + c
  const float* src = aug + p * (size_t)NU * NU * NC + u2c;

  float m = src[0];
  float s = 1.0f;
  for (int u1 = 1; u1 < NU; ++u1) {
    float v = src[(size_t)u1 * NU * NC];
    if (v <= m) {
      s += expf(v - m);
    } else {
      s = s * expf(m - v) + 1.0f;
      m = v;
    }
  }
  lseT[i] = m + logf(s);
}

// ---- Kernel 1: weights + folded per-class base ---------------------------
// w (23x24) Gumbel softmax; lse over class_logits; lse over U of feature
// tables staged into LDS; base[c] = cl[c]-lseC - sum_f fw[f]*lf[f][c] with
// fw[0]=1, fw[f]=w[f-1,f]. The fold is a (1x24)x(24x128) matvec done with
// V_WMMA_F32_16X16X4_F32 (A row 0 = weights, 6 K-steps x 8 N-tiles).
__global__ void __launch_bounds__(128)
prep_kernel(const float* __restrict__ cl, const float* __restrict__ featL,
            const float* __restrict__ slog, const float* __restrict__ nu,
            float* __restrict__ base, float* __restrict__ fw,
            float* __restrict__ pw) {
  __shared__ float w_s[NF - 1][NF];
  __shared__ float fw_s[NF];
  __shared__ float lf_s[NF][NC];
  const int t = threadIdx.x;

  if (t < NF - 1) {
    // Two-pass Gumbel softmax over valid cols j <= t+1 (no private arrays).
    float m = -1e30f;
    for (int j = 0; j <= t + 1; ++j) {
      float g = -logf(-logf(nu[t * NF + j] + EPSV) + EPSV);
      m = fmaxf(m, slog[t * NF + j] + g);
    }
    float s = 0.f;
    for (int j = 0; j < NF; ++j) {
      float e = 0.f;
      if (j <= t + 1) {
        float g = -logf(-logf(nu[t * NF + j] + EPSV) + EPSV);
        e = expf(slog[t * NF + j] + g - m);
      }
      w_s[t][j] = e;
      s += e;
    }
    float inv = 1.0f / s;
    for (int j = 0; j < NF; ++j) w_s[t][j] *= inv;
  }
  __syncthreads();
  if (t < NF) fw_s[t] = (t == 0) ? 1.0f : w_s[t - 1][t];

  // Per-class feature normalizers lf[f][c] (thread t == class c, coalesced).
  for (int f = 0; f < NF; ++f) {
    const float* col = featL + (size_t)f * NU * NC + t;
    float mm = -1e30f;
    for (int u = 0; u < NU; ++u) mm = fmaxf(mm, col[(size_t)u * NC]);
    float ss = 0.f;
    for (int u = 0; u < NU; ++u) ss += expf(col[(size_t)u * NC] - mm);
    lf_s[f][t] = mm + logf(ss);
  }

  // lse over class logits (tiny; every thread computes it).
  float m = -1e30f;
  for (int i = 0; i < NC; ++i) m = fmaxf(m, cl[i]);
  float s = 0.f;
  for (int i = 0; i < NC; ++i) s += expf(cl[i] - m);
  const float lse_c = m + logf(s);
  __syncthreads();

#if HAVE_WMMA
  if (t < 32) {  // wave 0, EXEC all-ones across its 32 lanes
    const int lane = t;
    const int mrow = lane & 15;   // M (A) / N (B,D) position
    const int khalf = lane >> 4;  // which K half this lane holds
    for (int T = 0; T < 8; ++T) { // 8 tiles of 16 classes
      v8f acc = {};
      for (int kk = 0; kk < 6; ++kk) {  // K = 24 in chunks of 4
        v2f a, bf;
        // A 16x4: lanes0-15 hold K=0,1 (vgpr0,1); lanes16-31 hold K=2,3.
        a.x = (mrow == 0) ? fw_s[kk * 4 + khalf * 2 + 0] : 0.f;
        a.y = (mrow == 0) ? fw_s[kk * 4 + khalf * 2 + 1] : 0.f;
        // B 4x16: vgpr0 rows K=0(lanes0-15)/K=1(lanes16-31), vgpr1 K=2/K=3.
        bf.x = lf_s[kk * 4 + khalf][T * 16 + mrow];
        bf.y = lf_s[kk * 4 + 2 + khalf][T * 16 + mrow];
        acc = __builtin_amdgcn_wmma_f32_16x16x4_f32(
            false, a, false, bf, (short)0, acc, false, false);
      }
      // D row M=0 lives in acc[0], lanes 0..15 (N = lane).
      if (lane < 16) {
        int c = T * 16 + lane;
        base[c] = cl[c] - lse_c - acc[0];
      }
    }
  }
#else
  {
    float bb = cl[t] - lse_c;
    for (int f = 0; f < NF; ++f) bb -= fw_s[f] * lf_s[f][t];
    base[t] = bb;
  }
#endif

  if (t < NF) fw[t] = fw_s[t];
  for (int p = t; p < NP; p += NC) {
    int f = 1;
    while ((f * (f + 1)) / 2 <= p) ++f;
    int j = p - (f * (f - 1)) / 2;
    pw[p] = w_s[f - 1][j];
  }
}

// ---- Host launcher -------------------------------------------------------
extern "C" void kernel_launch(void* const* d_in, const int* in_sizes, int n_in,
                              void* d_out, int out_size, void* d_ws,
                              size_t ws_size, hipStream_t stream) {
  const int* x      = (const int*)d_in[0];
  const float* cl   = (const float*)d_in[1];
  const float* featL= (const float*)d_in[2];
  const float* aug  = (const float*)d_in[3];
  const float* slog = (const float*)d_in[4];
  const float* nu   = (const float*)d_in[5];
  float* out = (float*)d_out;

  float* ws   = (float*)d_ws;
  float* base = ws;          // 128
  float* fw   = ws + 128;    // 32 (24 used)
  float* pw   = ws + 160;    // 276
  float* lseT = ws + 512;    // NP*NU*NC = 2,260,992 floats (~9 MB)

  const int Bn = in_sizes[0] / NF;

  hipLaunchKernelGGL(prep_kernel, dim3(1), dim3(128), 0, stream,
                     cl, featL, slog, nu, base, fw, pw);

  const size_t total = (size_t)NP * NU * NC;
  hipLaunchKernelGGL(lse_aug_kernel, dim3((unsigned)((total + 255) / 256)),
                     dim3(256), 0, stream, aug, lseT);

  hipLaunchKernelGGL(main_kernel, dim3(Bn), dim3(128), 0, stream,
                     x, featL, aug, base, fw, pw, lseT, out);
}